// PRESENTCipher_56942676410719
// MI455X (gfx1250) — compile-verified
//
#include <hip/hip_runtime.h>
#include <math.h>

#define ROUNDS 31
#define BLOCK 64
#define KEYSZ 80
#define ROWS_PER_BLK 8
#define THREADS 256

// -10 / ln(2): exp(-10*d*d) == exp2(C * d*d), maps to bare v_exp_f32
#define NEG10_LOG2E (-14.426950408889634f)
// 10 / ln(2): sigmoid(10*(x-0.5)) = 1/(1+exp2(POS10_LOG2E*(0.5-x)))
#define POS10_LOG2E (14.426950408889634f)

// SBOX values packed as nibbles: bits [4i+3:4i] = SBOX[i]
// SBOX = {12,5,6,11,9,0,10,13,3,14,15,8,4,7,1,2}
#define SBOX_NIB 0x21748FE3DA09B65CULL

__device__ __forceinline__ float fast_rcp(float x) {
    return __builtin_amdgcn_rcpf(x);   // v_rcp_f32
}

__device__ __forceinline__ float soft_xor(float x, float y) {
    float xs = fast_rcp(1.0f + exp2f(POS10_LOG2E * (0.5f - x)));
    float ys = fast_rcp(1.0f + exp2f(POS10_LOG2E * (0.5f - y)));
    float t1 = xs * (1.0f - ys);
    float t2 = (1.0f - xs) * ys;
    float v  = t1 + t2 - t1 * t2;
    return fminf(fmaxf(v, 0.0f), 1.0f);
}

// Softmax over i of exp(-10*(15x - i)^2), i = 0..15, dotted with SBOX/15.
// Terms at |15x - i| >= 1.5 are <= 1.7e-10 of the max term (>= exp(-2.5)),
// i.e. far below f32 epsilon of the sum -> a 3-wide window around
// round(15x), clamped to [0,15], is numerically identical in f32.
__device__ __forceinline__ float soft_sbox(float x) {
    float x15 = x * 15.0f;
    float mf  = rintf(x15);                                // v_rndne_f32
    float i0f = fminf(fmaxf(mf - 1.0f, 0.0f), 13.0f);      // window start
    int   i0  = (int)i0f;
    unsigned w = (unsigned)(SBOX_NIB >> (4 * i0));         // sb[i0..i0+2] in low 12 bits
    float den = 0.0f, num = 0.0f;
#pragma unroll
    for (int j = 0; j < 3; ++j) {
        float d = x15 - (i0f + (float)j);
        float e = exp2f(NEG10_LOG2E * d * d);              // == exp(-10*d*d)
        den += e;
        num += e * (float)((w >> (4 * j)) & 0xFu);
    }
    return num * (1.0f / 15.0f) * fast_rcp(den);
}

// One async b32 copy: global (64-bit addr) -> LDS (byte offset in low 32 bits
// of the generic pointer). Tracked with ASYNCcnt.
__device__ __forceinline__ void async_g2l_b32(const float* gptr, const float* lptr) {
    unsigned lds_off = (unsigned)(unsigned long long)lptr;
    unsigned long long gaddr = (unsigned long long)gptr;
    asm volatile("global_load_async_to_lds_b32 %0, %1, off"
                 :: "v"(lds_off), "v"(gaddr) : "memory");
}

__device__ __forceinline__ void wait_async0() {
    asm volatile("s_wait_asynccnt 0" ::: "memory");
}

__global__ __launch_bounds__(THREADS)
void present_soft_kernel(const float* __restrict__ pt,
                         const float* __restrict__ key,
                         float* __restrict__ out,
                         int batch) {
    __shared__ float sA[ROWS_PER_BLK * BLOCK];
    __shared__ float sB[ROWS_PER_BLK * BLOCK];
    __shared__ float sK[ROWS_PER_BLK * KEYSZ];

    const int tid      = threadIdx.x;
    const int blockRow = blockIdx.x * ROWS_PER_BLK;

    // ---- Stage inputs into LDS with gfx1250 async copies ----
    // Plaintext tile: 8 rows * 64 = 512 floats (2 per thread), coalesced.
    {
        const float* g0 = pt + (size_t)blockRow * BLOCK;
#pragma unroll
        for (int i = 0; i < 2; ++i) {
            int idx = tid + i * THREADS;                 // 0..511
            if (blockRow + (idx >> 6) < batch)
                async_g2l_b32(g0 + idx, &sA[idx]);
        }
        // Key tile: 8 rows * 80 = 640 floats (up to 3 per thread).
        const float* k0 = key + (size_t)blockRow * KEYSZ;
#pragma unroll
        for (int i = 0; i < 3; ++i) {
            int idx = tid + i * THREADS;                 // 0..767
            if (idx < ROWS_PER_BLK * KEYSZ && blockRow + (idx / KEYSZ) < batch)
                async_g2l_b32(k0 + idx, &sK[idx]);
        }
    }
    wait_async0();
    __syncthreads();

    const int c   = tid & (BLOCK - 1);       // column 0..63
    const int r0  = (tid >> 6) * 2;          // first of my 2 rows: 0,2,4,6
    // Scatter index: new[j] = y[PERM[j]]  <=>  nxt[inv[c]] = y[c],
    // PERM[i] = 16*(i%4)+i/4  =>  inv[c] = 4*(c%16) + c/16.
    const int inv = 4 * (c & 15) + (c >> 4);

    float* cur = sA;
    float* nxt = sB;

    for (int r = 0; r < ROUNDS; ++r) {
        // key rolled r times: k_r[i] = k_0[(i - r) mod 80]
        int kidx = c + KEYSZ - r;            // in [50, 143]
        if (kidx >= KEYSZ) kidx -= KEYSZ;
#pragma unroll
        for (int rr = 0; rr < 2; ++rr) {
            int row = r0 + rr;
            float x = cur[row * BLOCK + c];
            float k = sK[row * KEYSZ + kidx];
            float y = soft_sbox(soft_xor(x, k));
            nxt[row * BLOCK + inv] = y;      // permutation folded into scatter
        }
        __syncthreads();
        float* t = cur; cur = nxt; nxt = t;
    }

    // Final XOR with key rolled ROUNDS times.
    int kidx = c + KEYSZ - ROUNDS;           // in [49, 112]
    if (kidx >= KEYSZ) kidx -= KEYSZ;
#pragma unroll
    for (int rr = 0; rr < 2; ++rr) {
        int row = r0 + rr;
        if (blockRow + row < batch) {
            float x = cur[row * BLOCK + c];
            float k = sK[row * KEYSZ + kidx];
            out[(size_t)(blockRow + row) * BLOCK + c] = soft_xor(x, k);
        }
    }
}

extern "C" void kernel_launch(void* const* d_in, const int* in_sizes, int n_in,
                              void* d_out, int out_size, void* d_ws, size_t ws_size,
                              hipStream_t stream) {
    (void)n_in; (void)out_size; (void)d_ws; (void)ws_size;
    const float* pt  = (const float*)d_in[0];   // (BATCH, 64)
    const float* key = (const float*)d_in[1];   // (BATCH, 80)
    float* out = (float*)d_out;                 // (BATCH, 64)

    int batch = in_sizes[0] / BLOCK;            // 262144
    int grid  = (batch + ROWS_PER_BLK - 1) / ROWS_PER_BLK;

    present_soft_kernel<<<dim3(grid), dim3(THREADS), 0, stream>>>(pt, key, out, batch);
}